// KBCriterion_45552423141451
// MI455X (gfx1250) — compile-verified
//
#include <hip/hip_runtime.h>
#include <hip/hip_bf16.h>

#define B_DIM 128
#define N_DIM 1024
#define KCHUNK 256                   // K split: 4 chunks of 256
#define KCHUNKS (N_DIM / KCHUNK)
#define NT 32                        // N columns per block (two 16-wide WMMA tiles)
#define RP_STRIDE 48                 // float2 stride per K row-pair (32 data + 16 pad)
                                     // 96 dwords == 32 mod 64 banks -> conflict-free halves

typedef __attribute__((ext_vector_type(2))) float v2f;
typedef __attribute__((ext_vector_type(8))) float v8f;

// Workspace layout (floats):
//   ws[0]         : n_pos accumulator
//   ws[1]         : numerator accumulator  sum_{b,i,j} (p_j-p_i)^2 C_ij
//   ws[2..2+N)    : s[j] = rowsum_j + colsum_j of C

__global__ void kbc_init_kernel(float* ws) {
    int t = blockIdx.x * blockDim.x + threadIdx.x;
    if (t < 2) ws[t] = 0.0f;
}

// One block per index j: colsum_j (strided), rowsum_j (coalesced), positives of row j.
__global__ __launch_bounds__(256)
void kbc_sums_kernel(const float* __restrict__ C, float* __restrict__ ws) {
    __shared__ float red_s[256];
    __shared__ float red_c[256];
    const int j = blockIdx.x;
    const int t = threadIdx.x;
    float cs = 0.0f, rs = 0.0f, cnt = 0.0f;
    for (int i = t; i < N_DIM; i += 256) {
        cs += C[i * N_DIM + j];          // column j
        float v = C[j * N_DIM + i];      // row j
        rs += v;
        cnt += (v > 0.0f) ? 1.0f : 0.0f;
    }
    red_s[t] = cs + rs;
    red_c[t] = cnt;
    __syncthreads();
    for (int s = 128; s > 0; s >>= 1) {
        if (t < s) { red_s[t] += red_s[t + s]; red_c[t] += red_c[t + s]; }
        __syncthreads();
    }
    if (t == 0) {
        ws[2 + j] = red_s[0];
        atomicAdd(&ws[0], red_c[0]);
    }
}

// Split-K WMMA GEMM over Y = P (128 x 1024) * C (1024 x 1024), fused reduction.
// Grid: (N/32) x (K chunks). 256 threads = 8 waves; wave w owns M rows w*16..w*16+15
// and a 16x32 tile (two f32 accumulators -> each A fragment feeds 2 WMMAs).
//
// B staging is pre-swizzled into WMMA fragment order: LDS holds float2 pairs
// (C[k][n], C[k+1][n]) so each lane's B fragment is ONE contiguous ds_load_b64
// into an even-aligned VGPR pair (no v_mov reassembly in the hot loop); the two
// tiles' fragments fuse into a single ds_load_2addr_b64.
//
// Epilogue adds -2*Y_partial*p always, and p^2*s[n] only on chunk 0 so every
// (b, n) element is counted exactly once across the grid.
__global__ __launch_bounds__(256)
void kbc_wmma_kernel(const float* __restrict__ P, const float* __restrict__ C,
                     float* __restrict__ ws) {
    __shared__ float2 ldsB2[(KCHUNK / 2) * RP_STRIDE];   // 49152 bytes

    const int tileN = blockIdx.x * NT;
    const int k0    = blockIdx.y * KCHUNK;
    const int tid   = threadIdx.x;

    // Stage + swizzle: read two consecutive C rows (b128 each), interleave into
    // row-pair float2 fragments: ldsB2[rp*48 + c] = (C[k0+2rp][tileN+c], C[k0+2rp+1][tileN+c])
    const float4* C4 = reinterpret_cast<const float4*>(C);
    for (int u = tid; u < (KCHUNK / 2) * (NT / 4); u += 256) {
        const int rp = u >> 3;        // K row-pair
        const int c4 = u & 7;         // float4 column group within the 32-wide strip
        const float4 q0 = C4[(k0 + 2 * rp)     * (N_DIM / 4) + (tileN >> 2) + c4];
        const float4 q1 = C4[(k0 + 2 * rp + 1) * (N_DIM / 4) + (tileN >> 2) + c4];
        const float* f0 = reinterpret_cast<const float*>(&q0);
        const float* f1 = reinterpret_cast<const float*>(&q1);
#pragma unroll
        for (int j = 0; j < 4; ++j) {
            float2 pr; pr.x = f0[j]; pr.y = f1[j];
            ldsB2[rp * RP_STRIDE + c4 * 4 + j] = pr;
        }
    }
    __syncthreads();

    const int wave  = tid >> 5;
    const int lane  = tid & 31;
    const int half  = lane >> 4;      // 0: K=k,k+1   1: K=k+2,k+3
    const int l16   = lane & 15;
    const int tileM = wave * 16;
    const float* Arow = P + (tileM + l16) * N_DIM + k0;

    v8f acc0 = {};
    v8f acc1 = {};
#pragma unroll 2
    for (int r = 0; r < KCHUNK; r += 4) {
        const int ka = r + (half << 1);
        // A fragment: A[M = tileM+l16][k0+ka], [ka+1]  (contiguous -> global_load_b64)
        const float2 av = *reinterpret_cast<const float2*>(Arow + ka);
        v2f a; a.x = av.x; a.y = av.y;
        // B fragments: one contiguous float2 each (ds_load_b64 / fused 2addr_b64)
        const float2* bp = ldsB2 + ((r >> 1) + half) * RP_STRIDE + l16;
        const float2 bv0 = bp[0];     // columns tileN + l16
        const float2 bv1 = bp[16];    // columns tileN + 16 + l16
        v2f b0; b0.x = bv0.x; b0.y = bv0.y;
        v2f b1; b1.x = bv1.x; b1.y = bv1.y;
        acc0 = __builtin_amdgcn_wmma_f32_16x16x4_f32(
            false, a, false, b0, (short)0, acc0, false, false);
        acc1 = __builtin_amdgcn_wmma_f32_16x16x4_f32(
            false, a, false, b1, (short)0, acc1, false, false);
    }

    // Epilogue: acc VGPR g holds Y_partial[tileM + g + 8*half][n].
    const float* s = ws + 2;
    const int n0   = tileN + l16;
    const int n1   = n0 + 16;
    const float s0 = s[n0];
    const float s1 = s[n1];
    const bool first_chunk = (blockIdx.y == 0);
    float local = 0.0f;
#pragma unroll
    for (int g = 0; g < 8; ++g) {
        const int m    = tileM + g + (half << 3);
        const float p0 = P[m * N_DIM + n0];
        const float p1 = P[m * N_DIM + n1];
        local -= 2.0f * (acc0[g] * p0 + acc1[g] * p1);
        if (first_chunk)
            local += p0 * p0 * s0 + p1 * p1 * s1;
    }
    // wave32 butterfly reduction
    for (int off = 16; off > 0; off >>= 1)
        local += __shfl_xor(local, off, 32);
    if (lane == 0) atomicAdd(&ws[1], local);
}

__global__ void kbc_finalize_kernel(const float* __restrict__ ws,
                                    float* __restrict__ out) {
    out[0] = ws[1] / ((float)B_DIM * (ws[0] + 1e-8f));
}

extern "C" void kernel_launch(void* const* d_in, const int* in_sizes, int n_in,
                              void* d_out, int out_size, void* d_ws, size_t ws_size,
                              hipStream_t stream) {
    const float* probs = (const float*)d_in[0];   // [B, N] f32
    const float* co    = (const float*)d_in[1];   // [N, N] f32
    float* ws  = (float*)d_ws;
    float* out = (float*)d_out;

    kbc_init_kernel<<<1, 64, 0, stream>>>(ws);
    kbc_sums_kernel<<<N_DIM, 256, 0, stream>>>(co, ws);
    dim3 grid(N_DIM / NT, KCHUNKS);
    kbc_wmma_kernel<<<grid, 256, 0, stream>>>(probs, co, ws);
    kbc_finalize_kernel<<<1, 1, 0, stream>>>(ws, out);
}